// Pointnet2SSG_86517821212893
// MI455X (gfx1250) — compile-verified
//
#include <hip/hip_runtime.h>
#include <hip/hip_bf16.h>

#define B_      8
#define N_      8192
#define CIN     64
#define COUT    128
#define NPOINT_ 2048
#define NSAMPLE_ 32
#define RADIUS2 0.04f
#define ALPHA_  0.2f
#define BN_EPS_ 1e-5f

typedef float v2f __attribute__((ext_vector_type(2)));
typedef float v8f __attribute__((ext_vector_type(8)));

// ---------------------------------------------------------------------------
// 1) Furthest point sampling: one 1024-thread workgroup per batch.
//    Each thread owns 8 points entirely in VGPRs (px/py/pz/dist); the 2048
//    sequential steps are pure VALU + a (val,idx) argmax via shfl + 32-entry
//    LDS reduce. Winner's coords are published by the owning thread through
//    an unrolled register select (no dynamic indexing -> no scratch).
// ---------------------------------------------------------------------------
__global__ __launch_bounds__(1024) void fps_kernel(const float* __restrict__ xyz,
                                                   int* __restrict__ fps_idx) {
    __shared__ float swv[32];
    __shared__ int   swi[32];
    __shared__ float sl[3];
    __shared__ int   sbest;

    const int b    = blockIdx.x;
    const int tid  = threadIdx.x;
    const int lane = tid & 31, wid = tid >> 5;
    const float* bx = xyz + (size_t)b * N_ * 3;

    float px[8], py[8], pz[8], dist[8];
#pragma unroll
    for (int q = 0; q < 8; ++q) {
        const int i = tid + q * 1024;          // strided ownership, coalesced load
        px[q] = bx[i * 3 + 0];
        py[q] = bx[i * 3 + 1];
        pz[q] = bx[i * 3 + 2];
        dist[q] = 1e10f;
    }
    if (tid == 0) { sl[0] = px[0]; sl[1] = py[0]; sl[2] = pz[0]; }  // point 0
    int last = 0;
    __syncthreads();

    for (int it = 0; it < NPOINT_; ++it) {
        if (tid == 0) fps_idx[b * NPOINT_ + it] = last;
        const float lx = sl[0], ly = sl[1], lz = sl[2];

        float bv = -1.0f; int bi = 0;
#pragma unroll
        for (int q = 0; q < 8; ++q) {
            const float dx = px[q] - lx, dy = py[q] - ly, dz = pz[q] - lz;
            const float d  = dx * dx + dy * dy + dz * dz;
            const float nd = fminf(dist[q], d);
            dist[q] = nd;
            if (nd > bv) { bv = nd; bi = tid + q * 1024; }  // ascending idx -> first max
        }
        // wave32 reduction (first-max tie-break: lower index wins)
        for (int off = 16; off > 0; off >>= 1) {
            const float ov = __shfl_down(bv, off);
            const int   oi = __shfl_down(bi, off);
            if (ov > bv || (ov == bv && oi < bi)) { bv = ov; bi = oi; }
        }
        if (lane == 0) { swv[wid] = bv; swi[wid] = bi; }
        __syncthreads();
        if (wid == 0) {
            bv = swv[lane]; bi = swi[lane];
            for (int off = 16; off > 0; off >>= 1) {
                const float ov = __shfl_down(bv, off);
                const int   oi = __shfl_down(bi, off);
                if (ov > bv || (ov == bv && oi < bi)) { bv = ov; bi = oi; }
            }
            if (lane == 0) sbest = bi;
        }
        __syncthreads();
        last = sbest;
        if ((last & 1023) == tid) {           // owner publishes coords for next step
            const int p = last >> 10;
            float x = 0.f, y = 0.f, z = 0.f;
#pragma unroll
            for (int q = 0; q < 8; ++q)
                if (q == p) { x = px[q]; y = py[q]; z = pz[q]; }
            sl[0] = x; sl[1] = y; sl[2] = z;
        }
        __syncthreads();
    }
}

// ---------------------------------------------------------------------------
// 2) Ball query: one wave32 per center. Ballot + prefix popcount collects the
//    first NSAMPLE indices (ascending), pads with the first hit. Forward
//    stream prefetch (-> global_prefetch_b8) two chunks ahead.
// ---------------------------------------------------------------------------
__global__ __launch_bounds__(256) void ballq_kernel(const float* __restrict__ xyz,
                                                    const int* __restrict__ fps_idx,
                                                    int* __restrict__ idx) {
    const int gw   = (int)((blockIdx.x * blockDim.x + threadIdx.x) >> 5);
    const int lane = threadIdx.x & 31;
    if (gw >= B_ * NPOINT_) return;
    const int b = gw / NPOINT_, j = gw % NPOINT_;

    const float* bx = xyz + (size_t)b * N_ * 3;
    const int ci = fps_idx[b * NPOINT_ + j];
    const float cx = bx[ci * 3 + 0], cy = bx[ci * 3 + 1], cz = bx[ci * 3 + 2];

    int* out = idx + ((size_t)b * NPOINT_ + j) * NSAMPLE_;
    int cnt = 0;
    int first = 0;
    for (int base = 0; base < N_ && cnt < NSAMPLE_; base += 32) {
        __builtin_prefetch(&bx[(base + 64) * 3 + lane], 0, 0);
        const int k = base + lane;
        float dx = bx[k * 3 + 0] - cx;
        float dy = bx[k * 3 + 1] - cy;
        float dz = bx[k * 3 + 2] - cz;
        const bool pred = (dx * dx + dy * dy + dz * dz) < RADIUS2;
        const unsigned mask = (unsigned)__ballot(pred);     // wave32: low 32 bits
        if (cnt == 0 && mask != 0u)
            first = base + (__ffs(mask) - 1);               // uniform across wave
        const int prefix = __popc(mask & ((1u << lane) - 1u));
        const int pos = cnt + prefix;
        if (pred && pos < NSAMPLE_) out[pos] = k;
        cnt += __popc(mask);
    }
    if (cnt < NSAMPLE_ && lane >= cnt && lane < NSAMPLE_) out[lane] = first;
}

// ---------------------------------------------------------------------------
// 3) Gather + max-pool: one block per (b,center), 64 threads = channels.
// ---------------------------------------------------------------------------
__global__ __launch_bounds__(64) void pool_kernel(const float* __restrict__ feats,
                                                  const int* __restrict__ idx,
                                                  float* __restrict__ pooled) {
    __shared__ int sidx[NSAMPLE_];
    const int g = blockIdx.x;                 // b*NPOINT_ + j
    const int b = g / NPOINT_, j = g % NPOINT_;
    const int c = threadIdx.x;
    if (c < NSAMPLE_) sidx[c] = idx[(size_t)g * NSAMPLE_ + c];
    __syncthreads();
    const float* row = feats + ((size_t)b * CIN + c) * N_;
    float m = -INFINITY;
#pragma unroll
    for (int s = 0; s < NSAMPLE_; ++s) m = fmaxf(m, row[sidx[s]]);
    pooled[((size_t)b * CIN + c) * NPOINT_ + j] = m;
}

// ---------------------------------------------------------------------------
// 4) GEMM y[b] = W(128x64) * pooled[b](64x2048) with V_WMMA_F32_16X16X4_F32.
//    One wave per 16x16 tile, 16 k-steps. Exact fp32 math.
//    A layout: lane L -> M=L&15, K=(L>>4)*2 .. +1
//    B layout: lane L -> N=L&15, K=(L>>4)*2 .. +1
//    C/D     : VGPR v -> M = v + 8*(L>>4), N = L&15
// ---------------------------------------------------------------------------
__global__ __launch_bounds__(32) void gemm_kernel(const float* __restrict__ W,
                                                  const float* __restrict__ pooled,
                                                  float* __restrict__ y) {
    const int jt = blockIdx.x;      // N tile: 0..127
    const int ot = blockIdx.y;      // M tile: 0..7
    const int b  = blockIdx.z;      // batch
    const int lane = threadIdx.x;
    const int half = lane >> 4;
    const int l15  = lane & 15;
    const int o0 = ot * 16, j0 = jt * 16;

    const float* P = pooled + (size_t)b * CIN * NPOINT_;
    v8f acc = {};
#pragma unroll
    for (int k0 = 0; k0 < CIN; k0 += 4) {
        const int kh = k0 + half * 2;
        v2f a;
        a.x = W[(o0 + l15) * CIN + kh];
        a.y = W[(o0 + l15) * CIN + kh + 1];
        v2f bb;
        bb.x = P[(size_t)kh * NPOINT_ + j0 + l15];
        bb.y = P[(size_t)(kh + 1) * NPOINT_ + j0 + l15];
        acc = __builtin_amdgcn_wmma_f32_16x16x4_f32(
            /*neg_a=*/false, a, /*neg_b=*/false, bb,
            /*c_mod=*/(short)0, acc, /*reuse_a=*/false, /*reuse_b=*/false);
    }
    float* Y = y + (size_t)b * COUT * NPOINT_;
#pragma unroll
    for (int v = 0; v < 8; ++v)
        Y[(size_t)(o0 + v + 8 * half) * NPOINT_ + j0 + l15] = acc[v];
}

// ---------------------------------------------------------------------------
// 5) BatchNorm over (B, n) per channel + affine + LeakyReLU.
//    One block per channel: reduce 16384 values, then apply.
// ---------------------------------------------------------------------------
__global__ __launch_bounds__(256) void bn_kernel(const float* __restrict__ y,
                                                 const float* __restrict__ gamma,
                                                 const float* __restrict__ beta,
                                                 float* __restrict__ out) {
    __shared__ float ss[256], sq[256];
    const int o = blockIdx.x;
    const int tid = threadIdx.x;
    const int total = B_ * NPOINT_;
    float s = 0.f, q = 0.f;
    for (int i = tid; i < total; i += 256) {
        const int b = i >> 11, j = i & (NPOINT_ - 1);
        const float v = y[((size_t)b * COUT + o) * NPOINT_ + j];
        s += v; q += v * v;
    }
    ss[tid] = s; sq[tid] = q;
    __syncthreads();
    for (int off = 128; off > 0; off >>= 1) {
        if (tid < off) { ss[tid] += ss[tid + off]; sq[tid] += sq[tid + off]; }
        __syncthreads();
    }
    const float mean = ss[0] / (float)total;
    const float var  = sq[0] / (float)total - mean * mean;
    const float rstd = rsqrtf(var + BN_EPS_);
    const float g = gamma[o], be = beta[o];
    for (int i = tid; i < total; i += 256) {
        const int b = i >> 11, j = i & (NPOINT_ - 1);
        const size_t a = ((size_t)b * COUT + o) * NPOINT_ + j;
        const float v = (y[a] - mean) * rstd * g + be;
        out[a] = v >= 0.f ? v : ALPHA_ * v;
    }
}

extern "C" void kernel_launch(void* const* d_in, const int* in_sizes, int n_in,
                              void* d_out, int out_size, void* d_ws, size_t ws_size,
                              hipStream_t stream) {
    const float* xyz   = (const float*)d_in[0];   // [8,8192,3]
    const float* feats = (const float*)d_in[1];   // [8,64,8192]
    const float* W     = (const float*)d_in[2];   // [128,64]
    const float* gamma = (const float*)d_in[3];   // [128]
    const float* beta  = (const float*)d_in[4];   // [128]
    float* out = (float*)d_out;                   // [8,128,2048]

    char* ws = (char*)d_ws;
    int*   fps    = (int*)ws;                                             // 64 KB
    int*   idx    = (int*)(ws + (64u << 10));                             // 2 MB
    float* pooled = (float*)(ws + (64u << 10) + (2u << 20));              // 4 MB
    float* y      = (float*)(ws + (64u << 10) + (2u << 20) + (4u << 20)); // 8 MB

    fps_kernel<<<B_, 1024, 0, stream>>>(xyz, fps);
    ballq_kernel<<<(B_ * NPOINT_ * 32) / 256, 256, 0, stream>>>(xyz, fps, idx);
    pool_kernel<<<B_ * NPOINT_, 64, 0, stream>>>(feats, idx, pooled);
    dim3 gg(NPOINT_ / 16, COUT / 16, B_);
    gemm_kernel<<<gg, 32, 0, stream>>>(W, pooled, y);
    bn_kernel<<<COUT, 256, 0, stream>>>(y, gamma, beta, out);
}